// DifferentiableJPEG_61795989455353
// MI455X (gfx1250) — compile-verified
//
#include <hip/hip_runtime.h>

// Differentiable JPEG forward for MI455X (gfx1250, wave32).
// One wave = one 16x16 image tile = four 8x8 JPEG blocks (2x2).
// 2D DCT / IDCT as block-diagonal 16x16 matmuls on V_WMMA_F32_16X16X4_F32.

typedef __attribute__((ext_vector_type(2))) float f2;
typedef __attribute__((ext_vector_type(2))) float v2f;
typedef __attribute__((ext_vector_type(4))) float f4;
typedef __attribute__((ext_vector_type(8))) float v8f;

// Orthonormal 8x8 DCT-II matrix D8[r][c] (row-major).
__constant__ float c_d8[64] = {
  0.35355339f,  0.35355339f,  0.35355339f,  0.35355339f,  0.35355339f,  0.35355339f,  0.35355339f,  0.35355339f,
  0.49039264f,  0.41573481f,  0.27778512f,  0.09754516f, -0.09754516f, -0.27778512f, -0.41573481f, -0.49039264f,
  0.46193977f,  0.19134172f, -0.19134172f, -0.46193977f, -0.46193977f, -0.19134172f,  0.19134172f,  0.46193977f,
  0.41573481f, -0.09754516f, -0.49039264f, -0.27778512f,  0.27778512f,  0.49039264f,  0.09754516f, -0.41573481f,
  0.35355339f, -0.35355339f, -0.35355339f,  0.35355339f,  0.35355339f, -0.35355339f, -0.35355339f,  0.35355339f,
  0.27778512f, -0.49039264f,  0.09754516f,  0.41573481f, -0.41573481f, -0.09754516f,  0.49039264f, -0.27778512f,
  0.19134172f, -0.46193977f,  0.46193977f, -0.19134172f, -0.19134172f,  0.46193977f, -0.46193977f,  0.19134172f,
  0.09754516f, -0.27778512f,  0.41573481f, -0.49039264f,  0.49039264f, -0.41573481f,  0.27778512f, -0.09754516f
};

// Transposed DCT matrix: c_d8t[c*8 + r] = D8[r][c].
__constant__ float c_d8t[64] = {
  0.35355339f,  0.49039264f,  0.46193977f,  0.41573481f,  0.35355339f,  0.27778512f,  0.19134172f,  0.09754516f,
  0.35355339f,  0.41573481f,  0.19134172f, -0.09754516f, -0.35355339f, -0.49039264f, -0.46193977f, -0.27778512f,
  0.35355339f,  0.27778512f, -0.19134172f, -0.49039264f, -0.35355339f,  0.09754516f,  0.46193977f,  0.41573481f,
  0.35355339f,  0.09754516f, -0.46193977f, -0.27778512f,  0.35355339f,  0.41573481f, -0.19134172f, -0.49039264f,
  0.35355339f, -0.09754516f, -0.46193977f,  0.27778512f,  0.35355339f, -0.41573481f, -0.19134172f,  0.49039264f,
  0.35355339f, -0.27778512f, -0.19134172f,  0.49039264f, -0.35355339f, -0.09754516f,  0.46193977f, -0.41573481f,
  0.35355339f, -0.41573481f,  0.19134172f,  0.09754516f, -0.35355339f,  0.49039264f, -0.46193977f,  0.27778512f,
  0.35355339f, -0.49039264f,  0.46193977f, -0.41573481f,  0.35355339f, -0.27778512f,  0.19134172f, -0.09754516f
};

// JPEG quant table stored COLUMN-major: c_jqt[c*8 + r] = JQ[r][c].
__constant__ float c_jqt[64] = {
  16, 12, 14, 14, 18, 24, 49, 72,      // col 0
  11, 12, 13, 17, 22, 35, 64, 92,      // col 1
  10, 14, 16, 22, 37, 55, 78, 95,      // col 2
  16, 19, 24, 29, 56, 64, 87, 98,      // col 3
  24, 26, 40, 51, 68, 81,103,112,      // col 4
  40, 58, 57, 87,109,104,121,100,      // col 5
  51, 60, 69, 80,103,113,120,103,      // col 6
  61, 55, 56, 62, 77, 92,101, 99       // col 7
};

#define LSTRIDE 17   // 16 + 1 pad: conflict-free column (A-operand) reads

__device__ __forceinline__ v2f sel2(bool nz, f2 v) {
  v2f r; r.x = nz ? v.x : 0.0f; r.y = nz ? v.y : 0.0f; return r;
}

__global__ __launch_bounds__(256)
void diff_jpeg_kernel(const float* __restrict__ img, const int* __restrict__ qual,
                      float* __restrict__ out) {
  __shared__ float lds[8][16 * LSTRIDE];

  const int lane = threadIdx.x & 31;
  const int wid  = threadIdx.x >> 5;
  float* t = lds[wid];

  // tile index -> (image plane, tile y, tile x); 512x512 plane = 32x32 tiles
  const int tile   = blockIdx.x * 8 + wid;
  const int imgIdx = tile >> 10;
  const int tin    = tile & 1023;
  const int ty     = tin >> 5;
  const int tx     = tin & 31;
  const int base   = imgIdx * 262144 + (ty * 16) * 512 + tx * 16;

  // ---- global -> LDS, row-major, coalesced float4; apply x - 0.5 ----
  {
    const int r  = lane >> 1;
    const int c0 = (lane & 1) * 8;
    const float* src = img + base + r * 512 + c0;
    f4 p0 = *(const f4*)(src);
    f4 p1 = *(const f4*)(src + 4);
    float* dst = t + r * LSTRIDE + c0;
    dst[0] = p0.x - 0.5f; dst[1] = p0.y - 0.5f; dst[2] = p0.z - 0.5f; dst[3] = p0.w - 0.5f;
    dst[4] = p1.x - 0.5f; dst[5] = p1.y - 0.5f; dst[6] = p1.z - 0.5f; dst[7] = p1.w - 0.5f;
  }

  const int n  = lane & 15;   // column (B/C layouts) or row (A layout)
  const int h  = lane >> 4;   // lane half
  const int c8 = n & 7;

  // ---- per-lane constant operand chunks of D16 = diag(D8, D8) ----
  // Vector loads: one row and one column of D8, split at k&7 = {2h,2h+1} and {4+2h,4+2h+1}.
  const f2 rowLo = *(const f2*)(c_d8  + c8 * 8 + 2 * h);       // D8[c8][2h..]
  const f2 rowHi = *(const f2*)(c_d8  + c8 * 8 + 4 + 2 * h);   // D8[c8][4+2h..]
  const f2 colLo = *(const f2*)(c_d8t + c8 * 8 + 2 * h);       // D8[2h..][c8]
  const f2 colHi = *(const f2*)(c_d8t + c8 * 8 + 4 + 2 * h);   // D8[4+2h..][c8]
  const bool nz0 = (n < 8);   // lane's block row/col in upper-left D8
  // dRow[kc] = D16[n][4kc+v+2h]  (stage1 A-chunk == stage2 B-chunk)
  // dCol[kc] = D16[4kc+v+2h][n]  (stage3 A-chunk == stage4 B-chunk)
  v2f dRow[4], dCol[4];
  dRow[0] = sel2(nz0,  rowLo); dRow[1] = sel2(nz0,  rowHi);
  dRow[2] = sel2(!nz0, rowLo); dRow[3] = sel2(!nz0, rowHi);
  dCol[0] = sel2(nz0,  colLo); dCol[1] = sel2(nz0,  colHi);
  dCol[2] = sel2(!nz0, colLo); dCol[3] = sel2(!nz0, colHi);

  // ---- quant-table column this lane needs (q row = v, col = c8) ----
  const int quality = qual[0];
  const float sc = 0.01f * ((quality < 50) ? (5000.0f / (float)quality)
                                           : (200.0f - 2.0f * (float)quality));
  const f4 jq0 = *(const f4*)(c_jqt + c8 * 8);
  const f4 jq1 = *(const f4*)(c_jqt + c8 * 8 + 4);
  float q[8], rq[8];
#pragma unroll
  for (int v = 0; v < 8; ++v) {
    float jv = (v < 4) ? ((v == 0) ? jq0.x : (v == 1) ? jq0.y : (v == 2) ? jq0.z : jq0.w)
                       : ((v == 4) ? jq1.x : (v == 5) ? jq1.y : (v == 6) ? jq1.z : jq1.w);
    float qq = fminf(fmaxf(jv * sc, 1.0f), 255.0f);
    q[v] = qq;
    // reciprocal: v_rcp_f32 + one Newton step (~0.5 ulp)
    float r0 = __builtin_amdgcn_rcpf(qq);
    rq[v] = r0 * __builtin_fmaf(-qq, r0, 2.0f);
  }

  const int aRowOff = n * LSTRIDE + 2 * h;        // A-chunk base (row n, k start)
  const int bColOff = 2 * h * LSTRIDE + n;        // B-chunk base (k start, col n)
  const int cOff    = 8 * h * LSTRIDE + n;        // C/D layout base

  // ---- stage 1: Y = D16 @ X  (A = D16 const, B = X from LDS) ----
  v2f b0, b1, b2, b3;
  b0.x = t[bColOff +  0 * LSTRIDE]; b0.y = t[bColOff +  1 * LSTRIDE];
  b1.x = t[bColOff +  4 * LSTRIDE]; b1.y = t[bColOff +  5 * LSTRIDE];
  b2.x = t[bColOff +  8 * LSTRIDE]; b2.y = t[bColOff +  9 * LSTRIDE];
  b3.x = t[bColOff + 12 * LSTRIDE]; b3.y = t[bColOff + 13 * LSTRIDE];
  v8f acc = {0.f,0.f,0.f,0.f,0.f,0.f,0.f,0.f};
  acc = __builtin_amdgcn_wmma_f32_16x16x4_f32(false, dRow[0], false, b0, (short)0, acc, false, false);
  acc = __builtin_amdgcn_wmma_f32_16x16x4_f32(false, dRow[1], false, b1, (short)0, acc, false, false);
  acc = __builtin_amdgcn_wmma_f32_16x16x4_f32(false, dRow[2], false, b2, (short)0, acc, false, false);
  acc = __builtin_amdgcn_wmma_f32_16x16x4_f32(false, dRow[3], false, b3, (short)0, acc, false, false);
#pragma unroll
  for (int v = 0; v < 8; ++v) t[cOff + v * LSTRIDE] = acc[v];

  // ---- stage 2: S = Y @ D16^T  (A = Y from LDS, B = D16^T const) ----
  v2f a0 = *(const v2f*)(t + aRowOff +  0);
  v2f a1 = *(const v2f*)(t + aRowOff +  4);
  v2f a2 = *(const v2f*)(t + aRowOff +  8);
  v2f a3 = *(const v2f*)(t + aRowOff + 12);
  v8f acc2 = {0.f,0.f,0.f,0.f,0.f,0.f,0.f,0.f};
  acc2 = __builtin_amdgcn_wmma_f32_16x16x4_f32(false, a0, false, dRow[0], (short)0, acc2, false, false);
  acc2 = __builtin_amdgcn_wmma_f32_16x16x4_f32(false, a1, false, dRow[1], (short)0, acc2, false, false);
  acc2 = __builtin_amdgcn_wmma_f32_16x16x4_f32(false, a2, false, dRow[2], (short)0, acc2, false, false);
  acc2 = __builtin_amdgcn_wmma_f32_16x16x4_f32(false, a3, false, dRow[3], (short)0, acc2, false, false);

  // ---- quantize (round-to-nearest-even) / dequantize; write for stage 3 ----
#pragma unroll
  for (int v = 0; v < 8; ++v) {
    float dq = rintf(acc2[v] * rq[v]) * q[v];
    t[cOff + v * LSTRIDE] = dq;
  }

  // ---- stage 3: U = D16^T @ Sdeq  (A = D16^T const, B = Sdeq from LDS) ----
  b0.x = t[bColOff +  0 * LSTRIDE]; b0.y = t[bColOff +  1 * LSTRIDE];
  b1.x = t[bColOff +  4 * LSTRIDE]; b1.y = t[bColOff +  5 * LSTRIDE];
  b2.x = t[bColOff +  8 * LSTRIDE]; b2.y = t[bColOff +  9 * LSTRIDE];
  b3.x = t[bColOff + 12 * LSTRIDE]; b3.y = t[bColOff + 13 * LSTRIDE];
  v8f acc3 = {0.f,0.f,0.f,0.f,0.f,0.f,0.f,0.f};
  acc3 = __builtin_amdgcn_wmma_f32_16x16x4_f32(false, dCol[0], false, b0, (short)0, acc3, false, false);
  acc3 = __builtin_amdgcn_wmma_f32_16x16x4_f32(false, dCol[1], false, b1, (short)0, acc3, false, false);
  acc3 = __builtin_amdgcn_wmma_f32_16x16x4_f32(false, dCol[2], false, b2, (short)0, acc3, false, false);
  acc3 = __builtin_amdgcn_wmma_f32_16x16x4_f32(false, dCol[3], false, b3, (short)0, acc3, false, false);
#pragma unroll
  for (int v = 0; v < 8; ++v) t[cOff + v * LSTRIDE] = acc3[v];

  // ---- stage 4: R = U @ D16  (A = U from LDS, B = D16 const) ----
  a0 = *(const v2f*)(t + aRowOff +  0);
  a1 = *(const v2f*)(t + aRowOff +  4);
  a2 = *(const v2f*)(t + aRowOff +  8);
  a3 = *(const v2f*)(t + aRowOff + 12);
  v8f acc4 = {0.f,0.f,0.f,0.f,0.f,0.f,0.f,0.f};
  acc4 = __builtin_amdgcn_wmma_f32_16x16x4_f32(false, a0, false, dCol[0], (short)0, acc4, false, false);
  acc4 = __builtin_amdgcn_wmma_f32_16x16x4_f32(false, a1, false, dCol[1], (short)0, acc4, false, false);
  acc4 = __builtin_amdgcn_wmma_f32_16x16x4_f32(false, a2, false, dCol[2], (short)0, acc4, false, false);
  acc4 = __builtin_amdgcn_wmma_f32_16x16x4_f32(false, a3, false, dCol[3], (short)0, acc4, false, false);

  // ---- clip(R + 0.5, 0, 1), relayout through LDS, coalesced store ----
#pragma unroll
  for (int v = 0; v < 8; ++v) {
    float r = fminf(fmaxf(acc4[v] + 0.5f, 0.0f), 1.0f);
    t[cOff + v * LSTRIDE] = r;
  }
  {
    const int r  = lane >> 1;
    const int c0 = (lane & 1) * 8;
    const float* s = t + r * LSTRIDE + c0;
    f4 p0, p1;
    p0.x = s[0]; p0.y = s[1]; p0.z = s[2]; p0.w = s[3];
    p1.x = s[4]; p1.y = s[5]; p1.z = s[6]; p1.w = s[7];
    float* dst = out + base + r * 512 + c0;
    *(f4*)(dst)     = p0;
    *(f4*)(dst + 4) = p1;
  }
}

extern "C" void kernel_launch(void* const* d_in, const int* in_sizes, int n_in,
                              void* d_out, int out_size, void* d_ws, size_t ws_size,
                              hipStream_t stream) {
  const float* img  = (const float*)d_in[0];
  const int*   qual = (const int*)d_in[1];
  float*       out  = (float*)d_out;

  const int nElem  = in_sizes[0];     // 32*3*512*512 = 25,165,824
  const int tiles  = nElem / 256;     // one 16x16 tile per wave
  const int blocks = tiles / 8;       // 8 waves (256 threads) per block

  diff_jpeg_kernel<<<blocks, 256, 0, stream>>>(img, qual, out);
}